// DynFilter_55972013802282
// MI455X (gfx1250) — compile-verified
//
#include <hip/hip_runtime.h>
#include <hip/hip_bf16.h>

#define KH 5
#define KW 5
#define PADR 2
#define TILE_W 32
#define TILE_H 8
#define SHW (TILE_W + 2 * PADR)   // 36
#define SHH (TILE_H + 2 * PADR)   // 12
#define NC 3
#define SMEM_ELEMS (NC * SHH * SHW)  // 1296 floats ~ 5.2 KB

__global__ __launch_bounds__(256) void DynFilter_55972013802282_kernel(
    const float* __restrict__ x, const float* __restrict__ f,
    float* __restrict__ out, int H, int W)
{
    __shared__ float xs[SMEM_ELEMS];

    const int tx  = threadIdx.x;                 // 0..31
    const int ty  = threadIdx.y;                 // 0..7
    const int tid = ty * TILE_W + tx;            // 0..255
    const int b   = blockIdx.z;
    const int h0  = blockIdx.y * TILE_H;
    const int w0  = blockIdx.x * TILE_W;
    const size_t HW = (size_t)H * (size_t)W;

    // --- 1) zero-fill LDS tile (provides the zero-padded halo) -------------
    for (int idx = tid; idx < SMEM_ELEMS; idx += 256)
        xs[idx] = 0.0f;
    __syncthreads();  // zero stores complete in LDS before async writes land

    // --- 2) async global->LDS stage of x tile + halo (in-bounds lanes) ----
    for (int idx = tid; idx < SMEM_ELEMS; idx += 256) {
        const int c   = idx / (SHH * SHW);
        const int rem = idx % (SHH * SHW);
        const int r   = rem / SHW;
        const int col = rem % SHW;
        const int gh  = h0 + r   - PADR;
        const int gw  = w0 + col - PADR;
        if (gh >= 0 && gh < H && gw >= 0 && gw < W) {
            const float* gp = x + ((size_t)(b * NC + c)) * HW
                                + (size_t)gh * (size_t)W + (size_t)gw;
            unsigned lds_addr = (unsigned)(uintptr_t)(&xs[idx]); // low 32 bits of flat LDS ptr = LDS offset
            asm volatile("global_load_async_to_lds_b32 %0, %1, off"
                         :: "v"(lds_addr), "v"(gp) : "memory");
        }
    }
    // Wait this wave's ASYNCcnt, then barrier so all waves' tiles are visible.
    asm volatile("s_wait_asynccnt 0x0" ::: "memory");
    __syncthreads();

    // --- 3) stream per-pixel filter weights (read-once -> NT) + accumulate -
    const int h = h0 + ty;
    const int w = w0 + tx;
    const float* fp = f + ((size_t)b * (NC * KH * KW)) * HW
                        + (size_t)h * (size_t)W + (size_t)w;
    float acc = 0.0f;
#pragma unroll
    for (int c = 0; c < NC; ++c) {
#pragma unroll
        for (int i = 0; i < KH; ++i) {
#pragma unroll
            for (int j = 0; j < KW; ++j) {
                const int p = c * (KH * KW) + i * KW + j;
                const float fv = __builtin_nontemporal_load(fp + (size_t)p * HW);
                const float xv = xs[(c * SHH + (ty + i)) * SHW + (tx + j)];
                acc = fmaf(xv, fv, acc);
            }
        }
    }

    float* op = out + (size_t)b * HW + (size_t)h * (size_t)W + (size_t)w;
    __builtin_nontemporal_store(acc, op);
}

extern "C" void kernel_launch(void* const* d_in, const int* in_sizes, int n_in,
                              void* d_out, int out_size, void* d_ws, size_t ws_size,
                              hipStream_t stream) {
    const float* x = (const float*)d_in[0];   // [4, 3, 512, 512] f32
    const float* f = (const float*)d_in[1];   // [4, 75, 512, 512] f32
    float* out     = (float*)d_out;           // [4, 1, 512, 512] f32

    const int B = 4, H = 512, W = 512;
    dim3 block(TILE_W, TILE_H, 1);            // 256 threads = 8 waves (wave32)
    dim3 grid(W / TILE_W, H / TILE_H, B);     // 16 x 64 x 4 = 4096 blocks
    DynFilter_55972013802282_kernel<<<grid, block, 0, stream>>>(x, f, out, H, W);
}